// DKVMN_45775761440856
// MI455X (gfx1250) — compile-verified
//
#include <hip/hip_runtime.h>
#include <hip/hip_bf16.h>
#include <cstdint>

#define NUM_Q 13523
#define DIM_S 128
#define SIZE_M 64
#define BB 64
#define TT 200

typedef float v2f __attribute__((ext_vector_type(2)));
typedef float v8f __attribute__((ext_vector_type(8)));

// ---- CDNA5 async LDS->global store (ASYNCcnt-tracked) ----------------------
#if __has_builtin(__builtin_amdgcn_s_wait_asynccnt)
#define WAIT_ASYNCCNT(n) __builtin_amdgcn_s_wait_asynccnt(n)
#else
#define WAIT_ASYNCCNT(n) asm volatile("s_wait_asynccnt %0" :: "i"(n) : "memory")
#endif

__device__ __forceinline__ void async_store_b128_from_lds(const float* gaddr,
                                                          const float* ldsaddr) {
    unsigned long long ga = (unsigned long long)(uintptr_t)gaddr;
    // flat pointer to LDS: ADDR[31:0] is the LDS byte offset (ISA 10.2 aperture map)
    unsigned int la = (unsigned int)(uintptr_t)ldsaddr;
    asm volatile("global_store_async_from_lds_b128 %0, %1, off"
                 :: "v"(ga), "v"(la) : "memory");
}

template<int ACT> __device__ __forceinline__ float activate(float v) {
    if (ACT == 1) return 1.f / (1.f + __expf(-v));
    if (ACT == 2) return tanhf(v);
    return v;
}

// ---------------------------------------------------------------------------
// Gather k = k_emb[q], v = v_emb[q + NUM_Q*r]; 4 rows/block, float4 lanes
// ---------------------------------------------------------------------------
__global__ void gather_kernel(const int* __restrict__ q, const int* __restrict__ r,
                              const float* __restrict__ k_emb, const float* __restrict__ v_emb,
                              float* __restrict__ Kbuf, float* __restrict__ Vbuf) {
    const int sub  = threadIdx.x >> 5;
    const int lane = threadIdx.x & 31;
    const int row  = blockIdx.x * 4 + sub;
    const int qv = q[row];
    const int rv = r[row];
    const float4 kv = ((const float4*)(k_emb + (size_t)qv * DIM_S))[lane];
    const float4 vv = ((const float4*)(v_emb + ((size_t)qv + (size_t)NUM_Q * rv) * DIM_S))[lane];
    ((float4*)(Kbuf + (size_t)row * DIM_S))[lane] = kv;
    ((float4*)(Vbuf + (size_t)row * DIM_S))[lane] = vv;
}

// ---------------------------------------------------------------------------
// WMMA fp32 GEMM: Out[M,N] = act(X @ Wt^T + bias). X split at compile-time
// KSPLIT across Xa/Xb. Double-buffered operand chunks (16 loads in flight).
// One wave = one 16x16 tile; 4 waves/block = 64 rows.
// ---------------------------------------------------------------------------
template<int KDIM, int KSPLIT, int ACT>
__global__ void wmma_gemm_kernel(const float* __restrict__ Xa,
                                 const float* __restrict__ Xb,
                                 const float* __restrict__ Wt,
                                 const float* __restrict__ bias,
                                 float* __restrict__ Out, int Ncols) {
    const int lane  = threadIdx.x & 31;
    const int wave  = threadIdx.x >> 5;
    const int mtile = (blockIdx.x * 4 + wave) * 16;
    const int ntile = blockIdx.y * 16;
    const int half  = lane >> 4;
    const int mrow  = mtile + (lane & 15);
    const int nrow  = ntile + (lane & 15);
    const int koff  = half * 2;                 // K-pair within each 4-block

    const float* abase0 = Xa + (size_t)mrow * KSPLIT + koff;
    const float* abase1 = (KSPLIT < KDIM) ? (Xb + (size_t)mrow * (KDIM - KSPLIT) + koff)
                                          : abase0;
    const float* bbase  = Wt + (size_t)nrow * KDIM + koff;

    constexpr int NK  = KDIM / 4;               // k-steps (32 or 64)
    constexpr int CH  = 8;                      // k-steps per chunk
    constexpr int NCH = NK / CH;

    v2f ab[2][CH], bb[2][CH];
    v8f acc = {0.f, 0.f, 0.f, 0.f, 0.f, 0.f, 0.f, 0.f};

    auto load_chunk = [&](int c, int s) {
#pragma unroll
        for (int i = 0; i < CH; ++i) {
            const int k0 = (c * CH + i) * 4;    // compile-time after unroll
            const float* ap = (k0 < KSPLIT) ? (abase0 + k0)
                                            : (abase1 + (k0 - KSPLIT));
            ab[s][i] = *(const v2f*)ap;
            bb[s][i] = *(const v2f*)(bbase + k0);
        }
    };

    load_chunk(0, 0);
#pragma unroll
    for (int c = 0; c < NCH; ++c) {
        if (c + 1 < NCH) load_chunk(c + 1, (c + 1) & 1);
        const int s = c & 1;
#pragma unroll
        for (int i = 0; i < CH; ++i)
            acc = __builtin_amdgcn_wmma_f32_16x16x4_f32(
                false, ab[s][i], false, bb[s][i], (short)0, acc, false, false);
    }

    const float bn = bias ? bias[nrow] : 0.f;   // one output column per lane
    const int mbase = mtile + half * 8;
#pragma unroll
    for (int j = 0; j < 8; ++j)
        Out[(size_t)(mbase + j) * Ncols + nrow] = activate<ACT>(acc[j] + bn);
}

// ---------------------------------------------------------------------------
// Fused e/a GEMM: shares the A-operand (V) between two accumulators.
//   E = sigmoid(V @ eW^T + eb),  A = tanh(V @ aW^T + ab)
// ---------------------------------------------------------------------------
__global__ void wmma_gemm_ea_kernel(const float* __restrict__ V,
                                    const float* __restrict__ eW, const float* __restrict__ ebias,
                                    const float* __restrict__ aW, const float* __restrict__ abias,
                                    float* __restrict__ Eout, float* __restrict__ Aout) {
    constexpr int KDIM = DIM_S;
    const int lane  = threadIdx.x & 31;
    const int wave  = threadIdx.x >> 5;
    const int mtile = (blockIdx.x * 4 + wave) * 16;
    const int ntile = blockIdx.y * 16;
    const int half  = lane >> 4;
    const int mrow  = mtile + (lane & 15);
    const int nrow  = ntile + (lane & 15);
    const int koff  = half * 2;

    const float* abase = V  + (size_t)mrow * KDIM + koff;
    const float* ebase = eW + (size_t)nrow * KDIM + koff;
    const float* bbase = aW + (size_t)nrow * KDIM + koff;

    constexpr int NK  = KDIM / 4;               // 32
    constexpr int CH  = 4;
    constexpr int NCH = NK / CH;                // 8

    v2f av[2][CH], ev[2][CH], bv[2][CH];
    v8f accE = {0.f, 0.f, 0.f, 0.f, 0.f, 0.f, 0.f, 0.f};
    v8f accA = {0.f, 0.f, 0.f, 0.f, 0.f, 0.f, 0.f, 0.f};

    auto load_chunk = [&](int c, int s) {
#pragma unroll
        for (int i = 0; i < CH; ++i) {
            const int k0 = (c * CH + i) * 4;
            av[s][i] = *(const v2f*)(abase + k0);
            ev[s][i] = *(const v2f*)(ebase + k0);
            bv[s][i] = *(const v2f*)(bbase + k0);
        }
    };

    load_chunk(0, 0);
#pragma unroll
    for (int c = 0; c < NCH; ++c) {
        if (c + 1 < NCH) load_chunk(c + 1, (c + 1) & 1);
        const int s = c & 1;
#pragma unroll
        for (int i = 0; i < CH; ++i) {
            accE = __builtin_amdgcn_wmma_f32_16x16x4_f32(
                false, av[s][i], false, ev[s][i], (short)0, accE, false, false);
            accA = __builtin_amdgcn_wmma_f32_16x16x4_f32(
                false, av[s][i], false, bv[s][i], (short)0, accA, false, false);
        }
    }

    const float be = ebias[nrow];
    const float ba = abias[nrow];
    const int mbase = mtile + half * 8;
#pragma unroll
    for (int j = 0; j < 8; ++j) {
        Eout[(size_t)(mbase + j) * DIM_S + nrow] = activate<1>(accE[j] + be);
        Aout[(size_t)(mbase + j) * DIM_S + nrow] = activate<2>(accA[j] + ba);
    }
}

// ---------------------------------------------------------------------------
// In-place softmax over rows of 64
// ---------------------------------------------------------------------------
__global__ void softmax64_kernel(float* __restrict__ Wbuf) {
    const int row = blockIdx.x;
    const int m   = threadIdx.x;
    __shared__ float sh[64];
    float x = Wbuf[(size_t)row * SIZE_M + m];
    sh[m] = x; __syncthreads();
    for (int s = 32; s > 0; s >>= 1) { if (m < s) sh[m] = fmaxf(sh[m], sh[m + s]); __syncthreads(); }
    const float mx = sh[0]; __syncthreads();
    const float ex = __expf(x - mx);
    sh[m] = ex; __syncthreads();
    for (int s = 32; s > 0; s >>= 1) { if (m < s) sh[m] += sh[m + s]; __syncthreads(); }
    Wbuf[(size_t)row * SIZE_M + m] = ex / sh[0];
}

// ---------------------------------------------------------------------------
// Sequential memory update; one workgroup per batch element.
// Ping-pong 2x32KB Mv state in LDS (dynamic, 64.25 KB of 320 KB/WGP).
// Each step: fused read-out + update into buffer nxt, then the 32 KB tile is
// streamed to HBM with GLOBAL_STORE_ASYNC_FROM_LDS_B128 (ASYNCcnt). A buffer
// is only overwritten after s_wait_asynccnt(16)+barrier, which guarantees the
// store that read it (2 steps earlier; exactly 16 async ops/wave/step) drained.
// ---------------------------------------------------------------------------
__global__ void recurrence_kernel(const float* __restrict__ Wbuf,
                                  const float* __restrict__ Ebuf,
                                  const float* __restrict__ Abuf,
                                  const float* __restrict__ Mv0,
                                  float* __restrict__ Rbuf,
                                  float* __restrict__ MvOut) {
    extern __shared__ float smem[];
    // NOTE: no pointer-array initializer here (addrspacecast in a static
    // aggregate initializer breaks ld.lld); compute buffer bases arithmetically.
    float* wsh = smem + 2 * SIZE_M * DIM_S;
    const int b = blockIdx.x;
    const int d = threadIdx.x;                  // 0..127 (column)

    float* out0 = MvOut + (size_t)b * (TT + 1) * SIZE_M * DIM_S;
    for (int m = 0; m < SIZE_M; ++m)
        smem[m * DIM_S + d] = Mv0[(size_t)m * DIM_S + d];
    __syncthreads();
#pragma unroll
    for (int i = 0; i < 16; ++i)                // Mv[:,0] = Mv_init (async)
        async_store_b128_from_lds(out0 + (size_t)(i * DIM_S + d) * 4,
                                  smem + (i * DIM_S + d) * 4);

    for (int t = 0; t < TT; ++t) {
        const int cur = t & 1, nxt = cur ^ 1;
        float* Mcur = smem + cur * (SIZE_M * DIM_S);
        float* Mnxt = smem + nxt * (SIZE_M * DIM_S);
        const size_t row = (size_t)b * TT + t;
        if (d < SIZE_M) wsh[d] = Wbuf[row * SIZE_M + d];
        const float e = Ebuf[row * DIM_S + d];
        const float a = Abuf[row * DIM_S + d];
        WAIT_ASYNCCNT(16);                      // my stores older than last step done
        __syncthreads();                        // ... for every wave; wsh visible

        float rd = 0.f;
#pragma unroll 8
        for (int m = 0; m < SIZE_M; ++m) {
            const float wm  = wsh[m];
            const float old = Mcur[m * DIM_S + d];
            rd += wm * old;                     // read uses Mv_{t-1}
            Mnxt[m * DIM_S + d] = old * (1.f - wm * e) + wm * a;
        }
        Rbuf[row * DIM_S + d] = rd;
        __syncthreads();                        // tile nxt fully written

        float* outt = MvOut + ((size_t)b * (TT + 1) + (t + 1)) * SIZE_M * DIM_S;
#pragma unroll
        for (int i = 0; i < 16; ++i)            // stream tile out of LDS
            async_store_b128_from_lds(outt + (size_t)(i * DIM_S + d) * 4,
                                      Mnxt + (i * DIM_S + d) * 4);
    }
    WAIT_ASYNCCNT(0);
}

// ---------------------------------------------------------------------------
// p = sigmoid(F . p_W + p_b)
// ---------------------------------------------------------------------------
__global__ void p_kernel(const float* __restrict__ F, const float* __restrict__ pW,
                         const float* __restrict__ pb, float* __restrict__ pOut) {
    const int row = blockIdx.x;
    const int d   = threadIdx.x;
    __shared__ float sh[DIM_S];
    sh[d] = F[(size_t)row * DIM_S + d] * pW[d];
    __syncthreads();
    for (int s = 64; s > 0; s >>= 1) { if (d < s) sh[d] += sh[d + s]; __syncthreads(); }
    if (d == 0) pOut[row] = 1.f / (1.f + __expf(-(sh[0] + pb[0])));
}

// ---------------------------------------------------------------------------
extern "C" void kernel_launch(void* const* d_in, const int* in_sizes, int n_in,
                              void* d_out, int out_size, void* d_ws, size_t ws_size,
                              hipStream_t stream) {
    const int*   q     = (const int*)  d_in[0];
    const int*   r     = (const int*)  d_in[1];
    const float* k_emb = (const float*)d_in[2];
    const float* v_emb = (const float*)d_in[3];
    const float* Mk    = (const float*)d_in[4];
    const float* Mv0   = (const float*)d_in[5];
    const float* f_W   = (const float*)d_in[6];
    const float* f_b   = (const float*)d_in[7];
    const float* p_W   = (const float*)d_in[8];
    const float* p_b   = (const float*)d_in[9];
    const float* e_W   = (const float*)d_in[10];
    const float* e_b   = (const float*)d_in[11];
    const float* a_W   = (const float*)d_in[12];
    const float* a_b   = (const float*)d_in[13];

    const size_t BT = (size_t)BB * TT;          // 12800 rows
    float* out   = (float*)d_out;
    float* pOut  = out;                         // (B,T)
    float* MvOut = out + BT;                    // (B,T+1,64,128)

    float* ws   = (float*)d_ws;
    float* Kbuf = ws;                           // BT*128
    float* Vbuf = Kbuf + BT * DIM_S;            // BT*128
    float* Wbuf = Vbuf + BT * DIM_S;            // BT*64
    float* Ebuf = Wbuf + BT * SIZE_M;           // BT*128
    float* Abuf = Ebuf + BT * DIM_S;            // BT*128
    float* Rbuf = Abuf + BT * DIM_S;            // BT*128
    float* Fbuf = Ebuf;                         // reuse E after recurrence

    gather_kernel<<<(int)(BT / 4), DIM_S, 0, stream>>>(q, r, k_emb, v_emb, Kbuf, Vbuf);

    // w-logits: 12800 x 64, K=128
    wmma_gemm_kernel<DIM_S, DIM_S, 0><<<dim3(BT / 64, SIZE_M / 16), 128, 0, stream>>>(
        Kbuf, nullptr, Mk, nullptr, Wbuf, SIZE_M);
    softmax64_kernel<<<(int)BT, SIZE_M, 0, stream>>>(Wbuf);

    // fused e/a: 12800 x 128, K=128, shared A-operand
    wmma_gemm_ea_kernel<<<dim3(BT / 64, DIM_S / 16), 128, 0, stream>>>(
        Vbuf, e_W, e_b, a_W, a_b, Ebuf, Abuf);

    // sequential scan: 64 chains, LDS-resident state, async tile streaming
    const size_t lds_bytes = (2 * SIZE_M * DIM_S + SIZE_M) * sizeof(float);
    recurrence_kernel<<<BB, DIM_S, lds_bytes, stream>>>(Wbuf, Ebuf, Abuf, Mv0, Rbuf, MvOut);

    // f = tanh([read, k] @ f_W^T + f_b): 12800 x 128, K=256 split at 128
    wmma_gemm_kernel<2 * DIM_S, DIM_S, 2><<<dim3(BT / 64, DIM_S / 16), 128, 0, stream>>>(
        Rbuf, Kbuf, f_W, f_b, Fbuf, DIM_S);

    p_kernel<<<(int)BT, DIM_S, 0, stream>>>(Fbuf, p_W, p_b, pOut);
}